// LocalAttention_75926431858856
// MI455X (gfx1250) — compile-verified
//
#include <hip/hip_runtime.h>

typedef float v2f __attribute__((ext_vector_type(2)));
typedef float v8f __attribute__((ext_vector_type(8)));

// Problem constants (fixed by the reference)
constexpr int H    = 256;
constexpr int W    = 256;
constexpr int HW   = H * W;
constexpr int CIN  = 64;
constexpr int DYN  = 32;          // dynamic (value) channels: 0..31; static: 32..63
constexpr int CHW  = CIN * HW;    // per-batch input stride
constexpr int OCHW = DYN * HW;    // per-batch output stride

__global__ __launch_bounds__(128) void
LocalAttention_75926431858856_kernel(const float* __restrict__ x,
                                     const float* __restrict__ ker,   // [25]
                                     float* __restrict__ out) {
    const int wave = threadIdx.x >> 5;
    const int lane = threadIdx.x & 31;
    const int n    = lane & 15;      // pixel / matrix row-col index within tile
    const int half = lane >> 4;      // which half-wave (K-pair / M-half select)
    const int kp   = half * 2;       // K-pair base for A/B fragments (f32 16x16x4)

    const int b  = blockIdx.z;
    const int h  = blockIdx.y;
    const int w0 = blockIdx.x * 64 + wave * 16;   // this wave's 16-pixel tile

    const float* __restrict__ xb = x + (size_t)b * CHW;

    // Per-wave scratch: banded Gram rows (20 cols x 16 pixels x 5 row-offsets)
    __shared__ float G[4][5][20][16];
    // Per-wave softmax weights per pixel
    __shared__ float SM[4][16][25];

    //--------------------------------------------------------------------
    // Phase 1: scores via banded Gram WMMA.
    // B[k=c, n] = x_static[32+c, h, w0+n]  (shared across all 5 row offsets)
    //--------------------------------------------------------------------
    v2f Bs[8];
#pragma unroll
    for (int ch = 0; ch < 8; ++ch) {
        const float* p0 = xb + (size_t)(DYN + ch * 4 + kp) * HW + h * W;
        Bs[ch].x = p0[w0 + n];        // center row/col always in range
        Bs[ch].y = p0[HW + w0 + n];   // next static channel
    }

    // Branchless zero-pad: clamp index, always load, cndmask result.
    const int  colA1 = w0 - 2 + n;                 // Gram tile 1 column
    const int  colA2 = w0 + 14 + n;                // Gram tile 2 column
    const bool cok1  = (unsigned)colA1 < (unsigned)W;
    const bool cok2  = (unsigned)colA2 < (unsigned)W;
    const int  cc1   = cok1 ? colA1 : 0;
    const int  cc2   = cok2 ? colA2 : 0;

#pragma unroll
    for (int di = 0; di < 5; ++di) {
        const int  row = h + di - 2;
        const bool rok = (unsigned)row < (unsigned)H;
        const int  rr  = rok ? row : 0;
        const int  off1 = rr * W + cc1;
        const int  off2 = rr * W + cc2;
        const bool ok1  = rok && cok1;
        const bool ok2  = rok && cok2;

        v8f D1 = {};   // Gram cols w0-2 .. w0+13
        v8f D2 = {};   // Gram cols w0+14 .. w0+29 (only first 4 used)
#pragma unroll
        for (int ch = 0; ch < 8; ++ch) {
            const float* pc0 = xb + (size_t)(DYN + ch * 4 + kp) * HW;
            const float* pc1 = pc0 + HW;
            const float a1x = pc0[off1], a1y = pc1[off1];
            const float a2x = pc0[off2], a2y = pc1[off2];
            v2f A1, A2;
            A1.x = ok1 ? a1x : 0.0f;
            A1.y = ok1 ? a1y : 0.0f;
            A2.x = ok2 ? a2x : 0.0f;
            A2.y = ok2 ? a2y : 0.0f;
            D1 = __builtin_amdgcn_wmma_f32_16x16x4_f32(false, A1, false, Bs[ch],
                                                       (short)0, D1, false, false);
            D2 = __builtin_amdgcn_wmma_f32_16x16x4_f32(false, A2, false, Bs[ch],
                                                       (short)0, D2, false, false);
        }
        // D layout: lane -> column n = lane%16, rows m = v + half*8
#pragma unroll
        for (int v = 0; v < 8; ++v)
            G[wave][di][half * 8 + v][n] = D1[v];
        if (half == 0) {
#pragma unroll
            for (int v = 0; v < 4; ++v)
                G[wave][di][16 + v][n] = D2[v];
        }
    }
    __syncthreads();

    //--------------------------------------------------------------------
    // Phase 2: per-pixel softmax over 25 taps (lanes n and n+16 duplicate)
    // score(di,dj) for pixel n lives at G[di][n+dj][n]
    //--------------------------------------------------------------------
    float s[25];
#pragma unroll
    for (int k = 0; k < 25; ++k) {
        const int di = k / 5, dj = k % 5;
        s[k] = G[wave][di][n + dj][n] * (ker[k] * 0.2f);   // kernel[k]/KSIZE
    }
    float mx = s[0];
#pragma unroll
    for (int k = 1; k < 25; ++k) mx = fmaxf(mx, s[k]);
    float sum = 0.0f;
#pragma unroll
    for (int k = 0; k < 25; ++k) { s[k] = __expf(s[k] - mx); sum += s[k]; }
    const float inv = 1.0f / sum;
    if (half == 0) {
#pragma unroll
        for (int k = 0; k < 25; ++k) SM[wave][n][k] = s[k] * inv;
    }
    __syncthreads();

    //--------------------------------------------------------------------
    // Phase 3: out[c, n] = sum_{w'} x_dyn[c, h+di, w'] * Band[w', n]
    // A[m=c, k] = x_dyn[ctile*16+m, row, w0-2+k]; B[k, n] = softmax band.
    // K = 20 positions -> 5 chained K4 steps; accumulate over di in D.
    //--------------------------------------------------------------------
    // Hoist per-lane column clamps (loop-invariant over di)
    int  colK0[5], colK1[5];
    bool okc0[5], okc1[5];
#pragma unroll
    for (int ch = 0; ch < 5; ++ch) {
        const int k0 = ch * 4 + kp;
        const int c0 = w0 - 2 + k0;
        const int c1 = c0 + 1;
        okc0[ch] = (unsigned)c0 < (unsigned)W;
        okc1[ch] = (unsigned)c1 < (unsigned)W;
        colK0[ch] = okc0[ch] ? c0 : 0;
        colK1[ch] = okc1[ch] ? c1 : 0;
    }
    const float* pc0 = xb + (size_t)(n) * HW;        // channel = m (tile 0)
    const float* pc1 = xb + (size_t)(16 + n) * HW;   // channel = 16+m (tile 1)

    v8f O0 = {};   // channels 0..15
    v8f O1 = {};   // channels 16..31
#pragma unroll
    for (int di = 0; di < 5; ++di) {
        const int  row = h + di - 2;
        const bool rok = (unsigned)row < (unsigned)H;
        const int  rbase = (rok ? row : 0) * W;
#pragma unroll
        for (int ch = 0; ch < 5; ++ch) {
            const int kg0 = ch * 4 + kp;
            const int kg1 = kg0 + 1;
            const int d0  = kg0 - n;          // dj index for this lane's B elems
            const int d1  = kg1 - n;
            v2f Bf;
            Bf.x = (d0 >= 0 && d0 < 5) ? SM[wave][n][di * 5 + d0] : 0.0f;
            Bf.y = (d1 >= 0 && d1 < 5) ? SM[wave][n][di * 5 + d1] : 0.0f;

            const int off0 = rbase + colK0[ch];
            const int off1 = rbase + colK1[ch];
            const bool ok0 = rok && okc0[ch];
            const bool ok1 = rok && okc1[ch];
            const float a00 = pc0[off0], a01 = pc0[off1];
            const float a10 = pc1[off0], a11 = pc1[off1];
            v2f A0, A1;
            A0.x = ok0 ? a00 : 0.0f;
            A0.y = ok1 ? a01 : 0.0f;
            A1.x = ok0 ? a10 : 0.0f;
            A1.y = ok1 ? a11 : 0.0f;

            O0 = __builtin_amdgcn_wmma_f32_16x16x4_f32(false, A0, false, Bf,
                                                       (short)0, O0, false, false);
            O1 = __builtin_amdgcn_wmma_f32_16x16x4_f32(false, A1, false, Bf,
                                                       (short)0, O1, false, false);
        }
    }

    // D layout: lane -> pixel n = lane%16, channel rows m = v + half*8
    float* ob = out + (size_t)b * OCHW + (size_t)h * W + (w0 + n);
#pragma unroll
    for (int v = 0; v < 8; ++v) {
        ob[(size_t)(half * 8 + v) * HW]      = O0[v];
        ob[(size_t)(16 + half * 8 + v) * HW] = O1[v];
    }
}

extern "C" void kernel_launch(void* const* d_in, const int* in_sizes, int n_in,
                              void* d_out, int out_size, void* d_ws, size_t ws_size,
                              hipStream_t stream) {
    (void)in_sizes; (void)n_in; (void)out_size; (void)d_ws; (void)ws_size;
    const float* x   = (const float*)d_in[0];   // [4,64,256,256] fp32
    const float* ker = (const float*)d_in[1];   // [25,1] fp32
    float* out = (float*)d_out;                 // [4,32,256,256] fp32

    dim3 grid(W / 64, H, 4);   // 4 w-blocks x 256 rows x 4 batches
    dim3 block(128);           // 4 wave32, one 16-pixel tile per wave
    LocalAttention_75926431858856_kernel<<<grid, block, 0, stream>>>(x, ker, out);
}